// IterativeNN_36936718745860
// MI455X (gfx1250) — compile-verified
//
#include <hip/hip_runtime.h>
#include <hip/hip_bf16.h>
#include <stdint.h>

// Problem constants (match reference)
#define NUM_VARS 2048
#define LEAVES   (2 * NUM_VARS)      // 4096
#define LEVELS   12
#define WIDTH    4096
#define FANIN    4
#define TOTAL    (LEAVES + LEVELS * WIDTH)   // 53248 nodes
#define BLOCK    1024                 // 32 waves of wave32
#define SHMEM_BYTES ((size_t)TOTAL * sizeof(float))   // 208 KB < 320 KB/WGP

// ---- CDNA5 async global->LDS helpers -------------------------------------
typedef int v4i __attribute__((ext_vector_type(4)));
typedef __attribute__((address_space(1))) v4i*  gv4i_p;   // global, non-const
typedef __attribute__((address_space(3))) v4i*  lv4i_p;   // LDS
typedef __attribute__((address_space(3))) void* lvoid_p;

__device__ __forceinline__ void async_load_b128(const void* g, void* lds_generic) {
#if defined(__gfx1250__)
#if __has_builtin(__builtin_amdgcn_global_load_async_to_lds_b128)
    __builtin_amdgcn_global_load_async_to_lds_b128(
        (gv4i_p)(uintptr_t)g,                   // inttoptr -> AS(1) v4i*
        (lv4i_p)(lvoid_p)lds_generic,           // generic -> AS(3) v4i*
        /*offset=*/0, /*cpol=*/0);
#else
    uint32_t lds_addr = (uint32_t)(uintptr_t)(lvoid_p)lds_generic;
    uint64_t gaddr    = (uint64_t)(uintptr_t)g;
    asm volatile("global_load_async_to_lds_b128 %0, %1, off"
                 :: "v"(lds_addr), "v"(gaddr) : "memory");
#endif
#else
    (void)g; (void)lds_generic;
#endif
}

__device__ __forceinline__ void wait_asynccnt0() {
#if defined(__gfx1250__)
#if __has_builtin(__builtin_amdgcn_s_wait_asynccnt)
    __builtin_amdgcn_s_wait_asynccnt(0);
#else
    asm volatile("s_wait_asynccnt 0x0" ::: "memory");
#endif
#endif
}

// WGP-scope prefetch (scope 0): fills WGP-near cache + GL2. __builtin_prefetch
// can only reach SE/SYS scope, which skip the WGP cache entirely.
__device__ __forceinline__ void prefetch_wgp(const void* base, uint32_t voffset) {
#if defined(__gfx1250__)
    asm volatile("global_prefetch_b8 %0, %1"
                 :: "v"(voffset), "s"((uint64_t)(uintptr_t)base) : "memory");
#else
    (void)base; (void)voffset;
#endif
}

// ---- Kernel: one workgroup per batch element, full node buffer in LDS ----
__global__ __launch_bounds__(BLOCK, 1)
void circuit_eval_kernel(const float* __restrict__ x,
                         const int*   __restrict__ child_idx,
                         const int*   __restrict__ op_type,
                         float*       __restrict__ out)
{
    extern __shared__ float buf[];            // TOTAL floats, addrspace(3)
    const int b = blockIdx.x;
    const int t = threadIdx.x;

    // --- Stage leaves: buf[0..NUM_VARS) = x row, via async global->LDS ---
    const float* xrow = x + (size_t)b * NUM_VARS;
    if (t < NUM_VARS / 4) {                   // 512 x B128 transfers (waves 0..15)
        async_load_b128(xrow + 4 * t, &buf[4 * t]);
    }
    wait_asynccnt0();
    __syncthreads();

    // buf[NUM_VARS..LEAVES) = 1 - x
    #pragma unroll
    for (int k = 0; k < NUM_VARS / BLOCK; ++k) {
        int i = t + k * BLOCK;
        buf[NUM_VARS + i] = 1.0f - buf[i];
    }
    __syncthreads();

    // --- 12 levels, 4096 nodes each; 4 nodes per thread per level ---
    for (int l = 0; l < LEVELS; ++l) {
        const int4* __restrict__ idx =
            (const int4*)(child_idx + (size_t)l * WIDTH * FANIN);
        const int* __restrict__ op = op_type + (size_t)l * WIDTH;
        const int base = LEAVES + l * WIDTH;

        // Prefetch next level's index (64 KB) + op (16 KB) blocks into WGP$.
        if (l + 1 < LEVELS) {
            prefetch_wgp(child_idx + (size_t)(l + 1) * WIDTH * FANIN,
                         (uint32_t)t * 64u);              // 1024 threads x 64B
            if (t < (WIDTH * 4) / 64) {                   // 256 threads x 64B
                prefetch_wgp(op_type + (size_t)(l + 1) * WIDTH,
                             (uint32_t)t * 64u);
            }
        }

        #pragma unroll
        for (int k = 0; k < WIDTH / BLOCK; ++k) {    // 4 nodes / thread
            const int n  = t + k * BLOCK;
            const int4 c = idx[n];                   // coalesced 16B load (L2-hot)
            const float a0 = buf[c.x];               // ds_load gathers
            const float a1 = buf[c.y];
            const float a2 = buf[c.z];
            const float a3 = buf[c.w];
            const float prod = (a0 * a1) * (a2 * a3);
            const float sum  = (a0 + a1) + (a2 + a3);
            buf[base + n] = (op[n] == 0) ? prod : sum;
        }
        __syncthreads();   // level l complete before level l+1 gathers
    }

    if (t == 0) out[b] = buf[TOTAL - 1];     // root = last node
}

// ---- Host-side launch -----------------------------------------------------
extern "C" void kernel_launch(void* const* d_in, const int* in_sizes, int n_in,
                              void* d_out, int out_size, void* d_ws, size_t ws_size,
                              hipStream_t stream) {
    (void)n_in; (void)d_ws; (void)ws_size; (void)out_size;
    const float* x         = (const float*)d_in[0];
    const int*   child_idx = (const int*)d_in[1];
    const int*   op_type   = (const int*)d_in[2];
    float*       out       = (float*)d_out;

    const int B = in_sizes[0] / NUM_VARS;

    // Allow >64KB dynamic LDS (idempotent; safe w.r.t. graph capture).
    (void)hipFuncSetAttribute((const void*)&circuit_eval_kernel,
                              hipFuncAttributeMaxDynamicSharedMemorySize,
                              (int)SHMEM_BYTES);

    circuit_eval_kernel<<<dim3(B), dim3(BLOCK), SHMEM_BYTES, stream>>>(
        x, child_idx, op_type, out);
}